// NeuralMemoryModule_68461778698430
// MI455X (gfx1250) — compile-verified
//
#include <hip/hip_runtime.h>
#include <hip/hip_bf16.h>

// Problem dims (fixed by reference)
#define BB 4
#define TT 512
#define DD 768
#define KK 128
#define HH 256
#define VV 128

typedef __attribute__((ext_vector_type(2)))  float v2f;
typedef __attribute__((ext_vector_type(8)))  float v8f;

// ---------------------------------------------------------------------------
// Kernel 1: projections k/v/q = x @ {WK,WV,WQ} via v_wmma_f32_16x16x4_f32
// (full f32 precision, matching the reference numerics; the GEMM is tiny so
// the lower per-op FLOPs of the K=4 f32 WMMA is irrelevant).
//
// Block = 256 threads (8 waves). All waves in a block share one
// (matrix, n-tile) pair; the 768x16 B-strip is staged once into LDS (48 KB).
// Wave w computes m-tile (mgroup*8 + w). Grid = 3 mats * 8 ntiles * 16 groups
// = 384 blocks.
//
// A 16x4 f32 fragment (ISA 7.12.2): lane m = lane%16; lanes 0-15 hold K=0,1
// (VGPR0,1), lanes 16-31 hold K=2,3  -> contiguous float2 per lane.
// B 4x16 fragment: lane n = lane%16; lanes 0-15 hold K=0,1; lanes 16-31 K=2,3.
// D 16x16 f32: VGPR r -> row (hi*8 + r), col lane%16.
// ---------------------------------------------------------------------------
__global__ __launch_bounds__(256) void proj_wmma_kernel(
    const float* __restrict__ x,    // [2048, 768]
    const float* __restrict__ WK,   // [768, 128]
    const float* __restrict__ WV,
    const float* __restrict__ WQ,
    float* __restrict__ kout,       // [2048, 128]
    float* __restrict__ vout,
    float* __restrict__ qout)
{
    __shared__ float sB[DD * 16];            // 48 KB B-strip: [k][n]

    const int tid  = threadIdx.x;
    const int lane = tid & 31;
    const int wave = tid >> 5;

    const int mat    = blockIdx.x >> 7;      // 0..2
    const int rem    = blockIdx.x & 127;
    const int ntile  = rem >> 4;             // 0..7
    const int mgroup = rem & 15;             // 0..15

    const float* __restrict__ W = (mat == 0) ? WK : (mat == 1) ? WV : WQ;
    float* __restrict__ out     = (mat == 0) ? kout : (mat == 1) ? vout : qout;

    const int n0 = ntile * 16;

    // cooperative stage of the B strip: W[k][n0 + n] -> sB[k*16 + n]
    #pragma unroll 4
    for (int i = tid; i < DD * 16; i += 256)
        sB[i] = W[(size_t)(i >> 4) * KK + n0 + (i & 15)];
    __syncthreads();

    const int mtile = mgroup * 8 + wave;
    const int m0    = mtile * 16;
    const int lmod  = lane & 15;
    const int hi    = lane >> 4;             // half-wave: K offset 0 or 2
    const float* __restrict__ xr = x + (size_t)(m0 + lmod) * DD;

    v8f c = {};
    #pragma unroll 4
    for (int kk = 0; kk < DD; kk += 4) {
        const int ko = kk + hi * 2;
        // A: contiguous K pair from this lane's row (one b64 load)
        v2f a = *(const v2f*)(xr + ko);
        // B: two conflict-free LDS reads (banks n+c and n+c+32)
        v2f b;
        b.x = sB[ko * 16 + lmod];
        b.y = sB[(ko + 1) * 16 + lmod];
        c = __builtin_amdgcn_wmma_f32_16x16x4_f32(false, a, false, b,
                                                  (short)0, c, false, false);
    }

    #pragma unroll
    for (int r = 0; r < 8; ++r)
        out[(size_t)(m0 + hi * 8 + r) * KK + n0 + lmod] = c[r];
}

// ---------------------------------------------------------------------------
// Kernel 2: sequential Titans scan. One workgroup (256 thr, 8 waves) per batch
// element. Fast weights W1/W2/b1/b2 live in LDS (~262 KB of the WGP's 320 KB);
// momentum buffers live in VGPRs (thread t owns W1[:,t] and W2[hbase:, v]).
// ---------------------------------------------------------------------------
__global__ __launch_bounds__(256) void titans_scan_kernel(
    const float* __restrict__ kbuf,  // [B,T,K]
    const float* __restrict__ vbuf,  // [B,T,V]
    const float* __restrict__ qbuf,  // [B,T,K]
    const float* __restrict__ W1in, const float* __restrict__ b1in,
    const float* __restrict__ W2in, const float* __restrict__ b2in,
    const float* __restrict__ alpha_l, const float* __restrict__ theta_l,
    const float* __restrict__ eta_l,
    float* __restrict__ out)         // [B,T,V]
{
    extern __shared__ float smem[];
    float* sW1  = smem;              // [K][H] = 128*256
    float* sW2  = sW1 + KK * HH;     // [H][V] = 256*128
    float* sb1  = sW2 + HH * VV;     // [H]
    float* sb2  = sb1 + HH;          // [V]
    float* sk   = sb2 + VV;          // [K]
    float* sv   = sk + KK;           // [V]
    float* sq   = sv + VV;           // [K]
    float* sh   = sq + KK;           // [H]
    float* serr = sh + HH;           // [V]
    float* sdh  = serr + VV;         // [H]

    const int tid = threadIdx.x;
    const int b   = blockIdx.x;

    const float alpha = 1.f / (1.f + __expf(-alpha_l[0]));
    const float theta = 1.f / (1.f + __expf(-theta_l[0]));
    const float eta   = 1.f / (1.f + __expf(-eta_l[0]));
    const float omA   = 1.f - alpha;

    // ---- init fast weights in LDS (linear copies), momentum in VGPRs ----
    #pragma unroll 8
    for (int j = 0; j < KK; ++j) sW1[j * HH + tid] = W1in[j * HH + tid];
    #pragma unroll 8
    for (int j = 0; j < KK; ++j) sW2[j * HH + tid] = W2in[j * HH + tid];
    if (tid < HH) sb1[tid] = b1in[tid];
    if (tid < VV) sb2[tid] = b2in[tid];

    float S1m[128], S2m[128];        // momenta: dW1[j][tid], dW2[hbase+j][v]
    #pragma unroll
    for (int j = 0; j < 128; ++j) { S1m[j] = 0.f; S2m[j] = 0.f; }
    float Sb1 = 0.f, Sb2 = 0.f;
    __syncthreads();

    const int v     = tid & (VV - 1);
    const int hbase = (tid >> 7) * 128;

    const float* __restrict__ kb = kbuf + (size_t)b * TT * KK;
    const float* __restrict__ vb = vbuf + (size_t)b * TT * VV;
    const float* __restrict__ qb = qbuf + (size_t)b * TT * KK;

    for (int t = 0; t < TT; ++t) {
        // phase 0: stage token k/v/q; prefetch next token (global_prefetch_b8)
        if (tid < KK) {
            sk[tid] = kb[(size_t)t * KK + tid];
            sv[tid] = vb[(size_t)t * VV + tid];
            sq[tid] = qb[(size_t)t * KK + tid];
            if (t + 1 < TT) {
                __builtin_prefetch(&kb[(size_t)(t + 1) * KK + tid], 0, 1);
                __builtin_prefetch(&vb[(size_t)(t + 1) * VV + tid], 0, 1);
                __builtin_prefetch(&qb[(size_t)(t + 1) * KK + tid], 0, 1);
            }
        }
        __syncthreads();

        // phase 1: pre[h] = k . W1[:,h] + b1[h];  h = relu(pre)
        float pre = sb1[tid];
        #pragma unroll 8
        for (int k = 0; k < KK; ++k)
            pre = __fmaf_rn(sk[k], sW1[k * HH + tid], pre);
        sh[tid] = fmaxf(pre, 0.f);
        __syncthreads();

        // phase 2: err[v] = h . W2[:,v] + b2[v] - vt[v]
        if (tid < VV) {
            float e = sb2[tid] - sv[tid];
            #pragma unroll 8
            for (int h = 0; h < HH; ++h)
                e = __fmaf_rn(sh[h], sW2[h * VV + tid], e);
            serr[tid] = e;
        }
        __syncthreads();

        // phase 3: dh[h] = (err . W2[h,:]) * (pre>0), rotated to avoid
        //          the stride-128 LDS bank conflict on W2 row reads
        {
            float acc = 0.f;
            #pragma unroll 8
            for (int j = 0; j < VV; ++j) {
                const int vv = (j + tid) & (VV - 1);
                acc = __fmaf_rn(serr[vv], sW2[tid * VV + vv], acc);
            }
            sdh[tid] = (pre > 0.f) ? acc : 0.f;
        }
        __syncthreads();

        // phase 4: momentum + decayed weight updates (S in VGPRs, W in LDS)
        const float errv = serr[v];
        #pragma unroll
        for (int j = 0; j < 128; ++j) {
            const int h = hbase + j;
            const float g = sh[h] * errv;                    // dW2[h][v]
            S2m[j] = __fmaf_rn(eta, S2m[j], -theta * g);
            sW2[h * VV + v] = __fmaf_rn(omA, sW2[h * VV + v], S2m[j]);
        }
        const float dht = sdh[tid];
        #pragma unroll
        for (int j = 0; j < 128; ++j) {
            const float g = sk[j] * dht;                     // dW1[j][tid]
            S1m[j] = __fmaf_rn(eta, S1m[j], -theta * g);
            sW1[j * HH + tid] = __fmaf_rn(omA, sW1[j * HH + tid], S1m[j]);
        }
        Sb1 = __fmaf_rn(eta, Sb1, -theta * dht);
        sb1[tid] = __fmaf_rn(omA, sb1[tid], Sb1);
        if (tid < VV) {
            Sb2 = __fmaf_rn(eta, Sb2, -theta * serr[tid]);
            sb2[tid] = __fmaf_rn(omA, sb2[tid], Sb2);
        }
        __syncthreads();

        // phase 5: retrieval hidden hq[h] = relu(q . W1n[:,h] + b1n[h])
        float prq = sb1[tid];
        #pragma unroll 8
        for (int k = 0; k < KK; ++k)
            prq = __fmaf_rn(sq[k], sW1[k * HH + tid], prq);
        sh[tid] = fmaxf(prq, 0.f);
        __syncthreads();

        // phase 6: out[v] = hq . W2n[:,v] + b2n[v]
        if (tid < VV) {
            float o = sb2[tid];
            #pragma unroll 8
            for (int h = 0; h < HH; ++h)
                o = __fmaf_rn(sh[h], sW2[h * VV + tid], o);
            out[((size_t)b * TT + t) * VV + tid] = o;
        }
        __syncthreads();
    }
}

// ---------------------------------------------------------------------------
extern "C" void kernel_launch(void* const* d_in, const int* in_sizes, int n_in,
                              void* d_out, int out_size, void* d_ws, size_t ws_size,
                              hipStream_t stream)
{
    (void)in_sizes; (void)n_in; (void)out_size; (void)ws_size;

    const float* x   = (const float*)d_in[0];
    const float* WK  = (const float*)d_in[1];
    const float* WV  = (const float*)d_in[2];
    const float* WQ  = (const float*)d_in[3];
    const float* W1  = (const float*)d_in[4];
    const float* b1  = (const float*)d_in[5];
    const float* W2  = (const float*)d_in[6];
    const float* b2  = (const float*)d_in[7];
    const float* al  = (const float*)d_in[8];
    const float* th  = (const float*)d_in[9];
    const float* et  = (const float*)d_in[10];
    float* out = (float*)d_out;

    // workspace: k | v | q  (3 MB total)
    float* kbuf = (float*)d_ws;
    float* vbuf = kbuf + (size_t)BB * TT * KK;
    float* qbuf = vbuf + (size_t)BB * TT * VV;

    // 3 mats * 8 ntiles * 16 m-groups = 384 blocks, 8 waves each
    proj_wmma_kernel<<<384, 256, 0, stream>>>(x, WK, WV, WQ, kbuf, vbuf, qbuf);

    const size_t smem = (size_t)(KK * HH + HH * VV + HH + VV +
                                 KK + VV + KK + HH + VV + HH) * sizeof(float);
    (void)hipFuncSetAttribute(reinterpret_cast<const void*>(&titans_scan_kernel),
                              hipFuncAttributeMaxDynamicSharedMemorySize,
                              (int)smem);
    titans_scan_kernel<<<BB, 256, smem, stream>>>(kbuf, vbuf, qbuf,
                                                  W1, b1, W2, b2,
                                                  al, th, et, out);
}